// XTermFrequency_5471788335935
// MI455X (gfx1250) — compile-verified
//
#include <hip/hip_runtime.h>
#include <hip/hip_bf16.h>
#include <stdint.h>

#define VOCAB   50257
#define NSPLIT  4
#define VSPLIT  ((VOCAB + NSPLIT - 1) / NSPLIT)   // 12565 bins per split (~50.3 KB fp32)
#define SEQ     2048
#define BLOCK   512                               // 16 wave32 waves
#define NCHUNK  (SEQ / BLOCK)                     // 4 async-staged chunks per row

// Counted async-DMA fence: loads complete in order, so ASYNCcnt <= n guarantees
// the oldest (NCHUNK - n) issued transfers have landed in LDS.
#define WAIT_ASYNCCNT(n) asm volatile("s_wait_asynccnt " #n ::: "memory")

__global__ __launch_bounds__(BLOCK)
void XTermFrequency_hist_kernel(const long long* __restrict__ assign,
                                const float* __restrict__ ones_w,
                                float* __restrict__ out)
{
    __shared__ float    hist[VSPLIT];          // per-split vocab histogram, LDS-resident
    __shared__ unsigned stage[NCHUNK][BLOCK];  // 4-deep async-to-LDS staging pipeline
    __shared__ float    rowsum_s;

    const int row   = blockIdx.x;
    const int split = blockIdx.y;
    const int lo    = split * VSPLIT;
    const int hi    = (lo + VSPLIT < VOCAB) ? (lo + VSPLIT) : VOCAB;
    const int nbin  = hi - lo;
    const int tid   = threadIdx.x;

    // Zero the LDS histogram (output slice is written exactly once -> no global
    // zero-init pass over the 103MB output).
    for (int i = tid; i < VSPLIT; i += BLOCK) hist[i] = 0.0f;
    if (tid == 0) rowsum_s = 0.0f;
    __syncthreads();

    const long long* arow = assign + (size_t)row * SEQ;
    const float*     wrow = ones_w + (size_t)row * SEQ;

    // LDS byte offset of the staging array (generic->LDS offset via 32-bit trunc).
    const unsigned stage_base = (unsigned)(unsigned long long)(const void*)stage;
    const unsigned long long sbase = (unsigned long long)(uintptr_t)arow;

    // Issue ALL chunk DMAs up front (GVS mode: saddr = row base, per-lane vaddr =
    // byte offset of the int64 token; we grab only its low dword since V < 2^32).
    #pragma unroll
    for (int c = 0; c < NCHUNK; ++c) {
        const int t = c * BLOCK + tid;
        const unsigned dst  = stage_base + (unsigned)t * 4u;  // &stage[c][tid]
        const unsigned goff = (unsigned)t * 8u;
        asm volatile("global_load_async_to_lds_b32 %0, %1, %2 offset:0"
                     :: "v"(dst), "v"(goff), "s"(sbase)
                     : "memory");
    }

    // Overlap the weight loads with the in-flight DMAs.
    float w[NCHUNK];
    float lsum = 0.0f;
    #pragma unroll
    for (int c = 0; c < NCHUNK; ++c) {
        w[c] = wrow[c * BLOCK + tid];
        lsum += w[c];
    }

    // Drain the pipeline with counted waits; each lane reads only its own slot,
    // so no barriers are needed between chunks.
    #define PROCESS(c)                                              \
        do {                                                        \
            const int v = (int)stage[c][tid];                       \
            if (v >= lo && v < hi) atomicAdd(&hist[v - lo], w[c]);  \
        } while (0)

    WAIT_ASYNCCNT(3); PROCESS(0);
    WAIT_ASYNCCNT(2); PROCESS(1);
    WAIT_ASYNCCNT(1); PROCESS(2);
    WAIT_ASYNCCNT(0); PROCESS(3);
    #undef PROCESS

    // Wave32 shuffle reduction, then one ds_add_f32 per wave (exact: integral fp32).
    #pragma unroll
    for (int o = 16; o > 0; o >>= 1) lsum += __shfl_xor(lsum, o, 32);
    if ((tid & 31) == 0) atomicAdd(&rowsum_s, lsum);
    __syncthreads();

    const float inv  = 1.0f / rowsum_s;
    float*      orow = out + (size_t)row * VOCAB + lo;
    // Scalar b32 stores: slice base is only 4B-aligned when row + split is odd, and 32 lanes
    // x 4B is already a fully coalesced 128B line per wave.
    for (int i = tid; i < nbin; i += BLOCK)
        orow[i] = hist[i] * inv;
}

extern "C" void kernel_launch(void* const* d_in, const int* in_sizes, int n_in,
                              void* d_out, int out_size, void* d_ws, size_t ws_size,
                              hipStream_t stream) {
    const long long* assign = (const long long*)d_in[0];  // int64 assignments (B,S)
    const float*     ones_w = (const float*)d_in[1];      // fp32 ones (B,S)
    float*           out    = (float*)d_out;              // fp32 (B, VOCAB)

    const int B = in_sizes[0] / SEQ;                      // 512
    dim3 grid((unsigned)B, NSPLIT);
    XTermFrequency_hist_kernel<<<grid, BLOCK, 0, stream>>>(assign, ones_w, out);
}